// EmbeddingCorruptor_29171417875042
// MI455X (gfx1250) — compile-verified
//
#include <hip/hip_runtime.h>
#include <hip/hip_bf16.h>
#include <stdint.h>

// Problem constants from the reference.
#define TT 512
#define SS 16
#define QQ 4
#define VV 1024
#define HH 128
#define H4 (HH / 4)   // float4 per embedding row (32)

typedef float f4 __attribute__((ext_vector_type(4)));

// Address-space-qualified pointer element types for gfx1250 LDS paths.
typedef __attribute__((address_space(1))) int   as1_int;    // global ("__device__")
typedef __attribute__((address_space(3))) int   as3_int;    // LDS ("__shared__")
typedef __attribute__((address_space(3))) float as3_float;
typedef __attribute__((address_space(3))) f4    as3_f4;

// Stand-ins for jax.random.split(key(42), 6): kc, ks, kl, kq, ku, kr.
#define KC 0x8f1bbcdcu
#define KS 0x2545f491u
#define KL 0x9e3779b9u
#define KQ 0x85ebca6bu
#define KU 0xc2b2ae35u
#define KR 0x27d4eb2fu

__device__ __forceinline__ unsigned mix32(unsigned h) {
  h ^= h >> 16; h *= 0x7feb352du;
  h ^= h >> 15; h *= 0x846ca68bu;
  h ^= h >> 16;
  return h;
}
__device__ __forceinline__ unsigned rng(unsigned key, unsigned idx) {
  return mix32(key + idx * 0x9E3779B9u);
}
__device__ __forceinline__ float u01(unsigned h) {
  return (float)(h >> 8) * (1.0f / 16777216.0f);
}

// Returns: token index in [0,V) to gather, or -1 => use null_embed[q].
__device__ __forceinline__ int compute_sel(unsigned r, int t, int q, int tok) {
  const float plev[QQ] = {0.01f, 0.02f, 0.05f, 0.10f};
  bool corrupt = u01(rng(KC, r)) < plev[q];
  // 8 random spans (num_spans = T/64), each toggling a random level subset.
  #pragma unroll
  for (unsigned i = 0; i < 8u; ++i) {
    int start = (int)(rng(KS, i) & (TT - 1));
    int len   = 2 + (int)(rng(KL, i) % 7u);          // [2, 8]
    int end   = min(start + len, TT);
    bool qsel = (rng(KQ, i * 4u + (unsigned)q) & 1u) != 0u;
    corrupt = corrupt || ((t >= start) && (t < end) && qsel);
  }
  if (!corrupt) return tok;
  if (u01(rng(KU, r)) < 0.7f) return -1;              // 70% -> null embedding
  return (int)(rng(KR, r) & (VV - 1));                // 30% -> random token
}

// Stage null_embed (Q*H floats = 2 KB) into LDS, async path if available.
// Reads/writes of the shared array go through explicit AS(3) pointers so the
// consumer loads are ds_load_b128, not flat_load.
__device__ __forceinline__ void stage_null(const float* __restrict__ nullg,
                                           float* nullLds) {
#if defined(__gfx1250__) && \
    __has_builtin(__builtin_amdgcn_global_load_async_to_lds_b32) && \
    __has_builtin(__builtin_amdgcn_s_wait_asynccnt)
  for (int i = (int)threadIdx.x; i < QQ * HH; i += 256) {
    __builtin_amdgcn_global_load_async_to_lds_b32(
        (as1_int*)(nullg + i),      // global src
        (as3_int*)(nullLds + i),    // LDS dst
        0, 0);
  }
  __builtin_amdgcn_s_wait_asynccnt(0);
#else
  as3_float* dst = (as3_float*)nullLds;
  for (int i = (int)threadIdx.x; i < QQ * HH; i += 256) dst[i] = nullg[i];
#endif
  __syncthreads();
}

// ---------------------------------------------------------------------------
// Streaming pass (emitted first so the async/ds/NT ops show in the snippet):
// wave-per-row gather. Lane l moves float4 #l of the row: one
// global_load_b128 (L2-hot 2 MB table) + one NT global_store_b128.
// q = row & 3 is loop-invariant per wave (stride is a multiple of 4), so the
// LDS null row and weight base hoist out of the loop.
// ---------------------------------------------------------------------------
__global__ __launch_bounds__(256) void ec_gather(const f4* __restrict__ w4,
                                                 const float* __restrict__ nullg,
                                                 const int* __restrict__ sel,
                                                 f4* __restrict__ out4,
                                                 int nRows) {
  __shared__ __align__(16) float nullLds[QQ * HH];
  stage_null(nullg, nullLds);
  const as3_f4* nullLds4 = (const as3_f4*)nullLds;   // ds_load_b128 reads

  const int lane = (int)(threadIdx.x & 31u);
  int wid = (int)(blockIdx.x * 8u + (threadIdx.x >> 5));
  int nW  = (int)(gridDim.x * 8u);

  for (int row = wid; row < nRows; row += nW) {
    if (lane == 0) __builtin_prefetch(sel + row + nW, 0, 0);  // global_prefetch_b8
    int sv = sel[row];                 // uniform across the wave
    int q  = row & (QQ - 1);
    f4 v = (sv < 0) ? (f4)nullLds4[(q << 5) + lane]
                    : w4[((size_t)((q << 10) | sv) << 5) + (size_t)lane];
    // Output is written once and never re-read: NT keeps weights/sel in L2.
    __builtin_nontemporal_store(v, &out4[((size_t)row << 5) + (size_t)lane]);
  }
}

// ---------------------------------------------------------------------------
// Pass 1: one thread per output row; all RNG decided once, 4B selector out.
// ---------------------------------------------------------------------------
__global__ __launch_bounds__(256) void ec_selector(const int* __restrict__ x,
                                                   int* __restrict__ sel,
                                                   int nRows) {
  int r = (int)(blockIdx.x * 256u + threadIdx.x);
  if (r >= nRows) return;
  // r = b*T*S*Q + t*S*Q + s*Q + q ; S*Q = 64
  int t = (r >> 6) & (TT - 1);
  int q = r & (QQ - 1);
  sel[r] = compute_sel((unsigned)r, t, q, x[r]);
}

// Fallback if d_ws is too small: fused RNG (lane-redundant but uniform).
__global__ __launch_bounds__(256) void ec_fused(const int* __restrict__ x,
                                                const f4* __restrict__ w4,
                                                const float* __restrict__ nullg,
                                                f4* __restrict__ out4,
                                                int nRows) {
  __shared__ __align__(16) float nullLds[QQ * HH];
  stage_null(nullg, nullLds);
  const as3_f4* nullLds4 = (const as3_f4*)nullLds;

  const int lane = (int)(threadIdx.x & 31u);
  int wid = (int)(blockIdx.x * 8u + (threadIdx.x >> 5));
  int nW  = (int)(gridDim.x * 8u);

  for (int row = wid; row < nRows; row += nW) {
    int tok = x[row];
    int t   = (row >> 6) & (TT - 1);
    int q   = row & (QQ - 1);
    int sv  = compute_sel((unsigned)row, t, q, tok);
    f4 v = (sv < 0) ? (f4)nullLds4[(q << 5) + lane]
                    : w4[((size_t)((q << 10) | sv) << 5) + (size_t)lane];
    __builtin_nontemporal_store(v, &out4[((size_t)row << 5) + (size_t)lane]);
  }
}

extern "C" void kernel_launch(void* const* d_in, const int* in_sizes, int n_in,
                              void* d_out, int out_size, void* d_ws, size_t ws_size,
                              hipStream_t stream) {
  const int*   x  = (const int*)d_in[0];     // (B, T*S*Q) int32
  const float* w  = (const float*)d_in[1];   // (Q, V, H) f32
  const float* ne = (const float*)d_in[2];   // (Q, H) f32
  f4* out4 = (f4*)d_out;                     // (B, T*S*Q, H) f32

  const int nRows = in_sizes[0];             // B * T * S * Q = 524288

  // ~32K waves total -> deep latency hiding on ~1K SIMD32s, 16 rows/wave.
  const int gatherBlocks = 4096;

  if (ws_size >= (size_t)nRows * sizeof(int)) {
    int* sel = (int*)d_ws;
    ec_selector<<<(nRows + 255) / 256, 256, 0, stream>>>(x, sel, nRows);
    ec_gather<<<gatherBlocks, 256, 0, stream>>>((const f4*)w, ne, sel, out4, nRows);
  } else {
    ec_fused<<<4096, 256, 0, stream>>>(x, (const f4*)w, ne, out4, nRows);
  }
}